// WaveletLayer_14688788152864
// MI455X (gfx1250) — compile-verified
//
#include <hip/hip_runtime.h>

typedef __attribute__((ext_vector_type(2))) float v2f;
typedef __attribute__((ext_vector_type(8))) float v8f;
typedef int v4i __attribute__((vector_size(16)));   // matches async builtin pointee

#define NTHREADS 256
#define NWAVES   8
#define ROWS     4          // rows per workgroup

#define N0 4096             // input row length
#define M1 2051             // level-1 output length
#define M2 1029
#define M3 518
#define OUTC 533

// padded LDS row strides (floats). signal element j lives at buf[8+j].
// strides chosen == 12 mod 64 so rows land on different LDS banks.
#define S1 4172
#define S2 2124
#define S3 1100

#define NT1 65              // ceil(M1/32) tiles (32 output positions per tile)
#define NT2 33
#define NT3 17

#if defined(__has_builtin)
#  if __has_builtin(__builtin_amdgcn_global_load_async_to_lds_b128)
#    define USE_ASYNC_LDS 1
#  endif
#endif

// Correlation filters: out[m] = sum_k FILT[f][k] * xp[2m+k]
__constant__ float FILT[2][8] = {
  // cA: dec_lo reversed
  {  0.23037781330885523f,  0.7148465705525415f,   0.6308807679295904f,
    -0.02798376941698385f, -0.18703481171888114f,  0.030841381835986965f,
     0.032883011666982945f,-0.010597401784997278f },
  // cD: dec_hi reversed = (k odd ? -1 : +1) * dec_lo[k]
  { -0.010597401784997278f,-0.032883011666982945f, 0.030841381835986965f,
     0.18703481171888114f, -0.02798376941698385f, -0.6308807679295904f,
     0.7148465705525415f,  -0.23037781330885523f }
};

// order-preserving float<->uint map so ds_max_u32/ds_min_u32 give float max/min
__device__ __forceinline__ unsigned fkey(float x) {
  unsigned u = __float_as_uint(x);
  return (u & 0x80000000u) ? ~u : (u | 0x80000000u);
}
__device__ __forceinline__ float fkey_inv(unsigned k) {
  return __uint_as_float((k & 0x80000000u) ? (k & 0x7fffffffu) : ~k);
}

// half-sample symmetric pads + zeroed slack (slack must be 0: WMMA NaN-propagates
// even through zero band weights)
__device__ __forceinline__ void fill_pads(float* buf, int stride, int N, int tid) {
  const int slackR = stride - (8 + N + 7);
  const int per = 15 + slackR;
  for (int q = tid; q < ROWS * per; q += NTHREADS) {
    const int r = q / per, s = q % per;
    float* b = buf + r * stride;
    if (s < 2)       b[s] = 0.f;                              // left slack
    else if (s < 8)  b[s] = b[8 + 5 - (s - 2)];               // xp[i]=x[5-i]
    else if (s < 15) b[8 + N + (s - 8)] = b[8 + N - 1 - (s - 8)]; // right sym
    else             b[8 + N + 7 + (s - 15)] = 0.f;           // right slack
  }
}

// one 32-position tile: 6 accumulating f32 WMMAs + scatter (CHECK only on the
// single partial tail tile per level, so the hot loop is branch-free)
template<int LAST, int CHECK>
__device__ __forceinline__ void dwt_tile(int tbase, const float* abase, int colOff,
                                         const v2f* bfr, int fF, int pP, int hi,
                                         int Mout, float* __restrict__ xout, int sOut,
                                         float* __restrict__ gout, int row0,
                                         float (&ac_s)[ROWS], float (&ac_q)[ROWS],
                                         float (&ac_mx)[ROWS], float (&ac_mn)[ROWS]) {
  v8f acc = {};
  const int col0 = colOff + 2 * tbase;    // even -> 8B-aligned ds_load_b64
#pragma unroll
  for (int c = 0; c < 6; ++c) {
    v2f a = *(const v2f*)(abase + col0 + 4 * c);
    acc = __builtin_amdgcn_wmma_f32_16x16x4_f32(
        false, a, false, bfr[c], (short)0, acc, false, false);
  }
  union { v8f v; float s[8]; } u; u.v = acc;
  // D layout: vgpr v, lane l -> element [M = v + 8*(l>=16)][N = l&15]
  if (fF == 0) {                    // cA -> next level (or global cA3)
#pragma unroll
    for (int v = 0; v < 8; ++v) {
      const int gD = (v >> 2) + (hi << 1);
      const int rD = v & 3;
      const int pos = tbase + (gD << 3) + pP;
      if (!CHECK || pos < Mout) {
        if (LAST) gout[(size_t)(row0 + rD) * OUTC + pos] = u.s[v];
        else      xout[rD * sOut + 8 + pos] = u.s[v];
      }
    }
  } else {                          // cD -> per-lane running stats (dual-issue VALU)
#pragma unroll
    for (int v = 0; v < 8; ++v) {
      const int gD = (v >> 2) + (hi << 1);
      const int rD = v & 3;
      const int pos = tbase + (gD << 3) + pP;
      if (!CHECK || pos < Mout) {
        const float val = u.s[v];
        ac_s[rD]  += val;
        ac_q[rD]  += val * val;
        ac_mx[rD]  = fmaxf(ac_mx[rD], val);
        ac_mn[rD]  = fminf(ac_mn[rD], val);
      }
    }
  }
}

template<int LAST>
__device__ __forceinline__ void dwt_level(const float* __restrict__ xin, int sIn,
                                          float* __restrict__ xout, int sOut,
                                          int Mout, int ntiles, int lev,
                                          float* s_sum, unsigned* s_mm,
                                          float* __restrict__ gout, int row0,
                                          const v2f* bfr, int lane, int wave) {
  // A-fragment addressing (f32 16x16x4): lane -> M = lane&15, K = 2*(lane>>4)+{0,1}
  const int iM = lane & 15;
  const int gA = iM >> 2;           // position-group 0..3
  const int rA = iM & 3;            // row-in-tile
  const int kb = (lane >> 4) << 1;
  const int fF = (lane >> 3) & 1;   // 0 = cA lanes, 1 = cD lanes (N = lane&15)
  const int pP = lane & 7;
  const int hi = lane >> 4;
  const float* abase = xin + rA * sIn;
  const int colOff = 2 + (gA << 4) + kb;

  const float INF = __builtin_inff();
  float ac_s[ROWS]  = {0.f, 0.f, 0.f, 0.f};
  float ac_q[ROWS]  = {0.f, 0.f, 0.f, 0.f};
  float ac_mx[ROWS] = {-INF, -INF, -INF, -INF};
  float ac_mn[ROWS] = { INF,  INF,  INF,  INF};

  const int nfull = ntiles - 1;     // last tile of every level is partial
  for (int t = wave; t < nfull; t += NWAVES)
    dwt_tile<LAST, 0>(t << 5, abase, colOff, bfr, fF, pP, hi, Mout,
                      xout, sOut, gout, row0, ac_s, ac_q, ac_mx, ac_mn);
  if (wave == (nfull & (NWAVES - 1)))   // wave-uniform branch: EXEC still all-1s
    dwt_tile<LAST, 1>(nfull << 5, abase, colOff, bfr, fF, pP, hi, Mout,
                      xout, sOut, gout, row0, ac_s, ac_q, ac_mx, ac_mn);

  // cross-lane butterfly over the 16 active cD lanes (masks stay inside the
  // active set: {8..15}x{24..31}), then one atomic per (row, stat)
  if (fF == 1) {
#pragma unroll
    for (int r = 0; r < ROWS; ++r) {
      float s = ac_s[r], q = ac_q[r], mx = ac_mx[r], mn = ac_mn[r];
#pragma unroll
      for (int m = 0; m < 4; ++m) {
        const int msk = (m == 3) ? 16 : (1 << m);
        s  += __shfl_xor(s,  msk, 32);
        q  += __shfl_xor(q,  msk, 32);
        mx  = fmaxf(mx, __shfl_xor(mx, msk, 32));
        mn  = fminf(mn, __shfl_xor(mn, msk, 32));
      }
      if (lane == 8) {
        float* ss = s_sum + (lev * ROWS + r) * 2;
        atomicAdd(ss,     s);                 // ds_add_f32 (8 waves contend)
        atomicAdd(ss + 1, q);
        unsigned* mm = s_mm + (lev * ROWS + r) * 2;
        atomicMax(mm,     fkey(mx));          // ds_max_u32
        atomicMin(mm + 1, fkey(mn));          // ds_min_u32
      }
    }
  }
}

__global__ __launch_bounds__(NTHREADS)
void dwt_db4_feats(const float* __restrict__ x, float* __restrict__ out) {
  extern __shared__ float smem[];
  float*    XP1   = smem;
  float*    XP2   = XP1 + ROWS * S1;
  float*    XP3   = XP2 + ROWS * S2;
  float*    s_sum = XP3 + ROWS * S3;                     // [3][ROWS][2] sum,sumsq
  unsigned* s_mm  = (unsigned*)(s_sum + 3 * ROWS * 2);   // [3][ROWS][2] maxkey,minkey

  const int tid  = threadIdx.x;
  const int lane = tid & 31;      // wave32
  const int wave = tid >> 5;
  const int row0 = blockIdx.x * ROWS;

  // ---- per-lane constant band-matrix B fragments (B[k][n], n = f*8+p) ----
  v2f bfr[6];
  {
    const int n = lane & 15;
    const int f = n >> 3;
    const int p = n & 7;
    const int kbB = (lane >> 4) << 1;
#pragma unroll
    for (int c = 0; c < 6; ++c) {
      const int t0 = 4 * c + kbB - 2 * p;
      const int t1 = t0 + 1;
      v2f bb;
      bb.x = (t0 >= 0 && t0 < 8) ? FILT[f][t0] : 0.f;
      bb.y = (t1 >= 0 && t1 < 8) ? FILT[f][t1] : 0.f;
      bfr[c] = bb;
    }
  }

  // ---- stage 4 rows into LDS; async direct-to-LDS when available ----
  {
    const float4* xg = (const float4*)(x + (size_t)row0 * N0);
    for (int q = tid; q < ROWS * (N0 / 4); q += NTHREADS) {
      const int r  = q >> 10;                 // N0/4 = 1024 float4 per row
      const int c4 = q & 1023;
      float* dst = XP1 + r * S1 + 8 + 4 * c4; // 16B-aligned LDS address
#ifdef USE_ASYNC_LDS
      __builtin_amdgcn_global_load_async_to_lds_b128(
          (__attribute__((address_space(1))) v4i*)(xg + r * (N0 / 4) + c4),
          (__attribute__((address_space(3))) v4i*)dst, 0, 0);
#else
      const float4 d = xg[r * (N0 / 4) + c4];
      dst[0] = d.x; dst[1] = d.y; dst[2] = d.z; dst[3] = d.w;
#endif
    }
    for (int q = tid; q < 3 * ROWS * 2; q += NTHREADS) {
      s_sum[q] = 0.f;
      s_mm[q]  = (q & 1) ? 0xffffffffu : 0u;  // min-slot : max-slot init
    }
  }
#ifdef USE_ASYNC_LDS
#  if __has_builtin(__builtin_amdgcn_s_wait_asynccnt)
  __builtin_amdgcn_s_wait_asynccnt(0);
#  else
  asm volatile("s_wait_asynccnt 0" ::: "memory");
#  endif
#endif
  __syncthreads();
  fill_pads(XP1, S1, N0, tid);
  __syncthreads();
  dwt_level<0>(XP1, S1, XP2, S2, M1, NT1, 0, s_sum, s_mm, out, row0, bfr, lane, wave);
  __syncthreads();
  fill_pads(XP2, S2, M1, tid);
  __syncthreads();
  dwt_level<0>(XP2, S2, XP3, S3, M2, NT2, 1, s_sum, s_mm, out, row0, bfr, lane, wave);
  __syncthreads();
  fill_pads(XP3, S3, M2, tid);
  __syncthreads();
  dwt_level<1>(XP3, S3, nullptr, 0, M3, NT3, 2, s_sum, s_mm, out, row0, bfr, lane, wave);
  __syncthreads();

  // ---- finalize stats: [mean, std, var, max, min]; cD1->528, cD2->523, cD3->518
  for (int q = tid; q < 3 * ROWS; q += NTHREADS) {
    const int lev = q % 3;
    const int r   = q / 3;
    const float cnt[3] = { (float)M1, (float)M2, (float)M3 };
    const float sum = s_sum[(lev * ROWS + r) * 2];
    const float sq  = s_sum[(lev * ROWS + r) * 2 + 1];
    const float inv = 1.f / cnt[lev];
    const float mean = sum * inv;
    const float var  = fmaxf(sq * inv - mean * mean, 0.f);
    float* o = out + (size_t)(row0 + r) * OUTC + (528 - 5 * lev);
    o[0] = mean;
    o[1] = sqrtf(var);
    o[2] = var;
    o[3] = fkey_inv(s_mm[(lev * ROWS + r) * 2]);
    o[4] = fkey_inv(s_mm[(lev * ROWS + r) * 2 + 1]);
  }
}

extern "C" void kernel_launch(void* const* d_in, const int* in_sizes, int n_in,
                              void* d_out, int out_size, void* d_ws, size_t ws_size,
                              hipStream_t stream) {
  (void)n_in; (void)out_size; (void)d_ws; (void)ws_size;
  const float* x = (const float*)d_in[0];
  float* out = (float*)d_out;
  const int nrows = in_sizes[0] / N0;                 // 16384
  const size_t shmem = (size_t)(ROWS * (S1 + S2 + S3) + 3 * ROWS * 2 * 2) * sizeof(float); // ~118.5 KB
  (void)hipFuncSetAttribute((const void*)dwt_db4_feats,
                            hipFuncAttributeMaxDynamicSharedMemorySize, (int)shmem);
  dwt_db4_feats<<<dim3(nrows / ROWS), dim3(NTHREADS), shmem, stream>>>(x, out);
}